// TreeLayerNorm_69157563400282
// MI455X (gfx1250) — compile-verified
//
#include <hip/hip_runtime.h>
#include <hip/hip_bf16.h>

// TreeLayerNorm for MI455X (gfx1250, wave32).
// B=64, C=64, L=8192. Memory-bound: ~325MB traffic -> ~14us floor @23.3TB/s.
// Cross-lane reductions use V_WMMA_F32_16X16X4_F32 with a ones-B matrix:
// D = A x ones => every D element is a row-sum of A; summing the 8 C/D VGPRs
// per lane + lo/hi half combine reduces both per-lane partials in one matrix op.

#define Bn   64
#define Cn   64
#define Ln   8192
#define SPLIT 16            // chunks (blocks) per batch in the reduce pass
#define TPB  256            // 8 waves of 32
#define NWAVES (TPB / 32)

static_assert((Cn * Ln) / 4 == (1 << 17), "b = i4 >> 17 relies on C*L/4 == 2^17");
static_assert(Cn % SPLIT == 0, "each chunk takes C/SPLIT whole rows");

typedef __attribute__((ext_vector_type(2))) float v2f;
typedef __attribute__((ext_vector_type(8))) float v8f;

// Reduce two per-lane partials across the wave with one WMMA.
// Writes the low-half total from lane 0 and the high-half total from lane 16
// into smem[2*wid] / smem[2*wid+1] (no shuffle needed; layout-robust).
__device__ __forceinline__ void wave_reduce_wmma(float p0, float p1,
                                                 float* smem2, int lane, int wid) {
    v2f a;    a.x = p0;   a.y = p1;     // 16x4 A matrix: 2 VGPRs, any placement OK
    v2f ones; ones.x = 1.0f; ones.y = 1.0f;  // 4x16 B = all ones
    v8f c = {};
    // emits v_wmma_f32_16x16x4_f32 (8-arg form: neg_a, A, neg_b, B, c_mod, C, reuse_a, reuse_b)
    v8f d = __builtin_amdgcn_wmma_f32_16x16x4_f32(
        false, a, false, ones, (short)0, c, false, false);
    // Per lane: sum the 8 accumulator VGPRs.
    // Lanes 0-15 hold sum of rows 0-7 (replicated across N); lanes 16-31 rows 8-15.
    float s = d[0] + d[1] + d[2] + d[3] + d[4] + d[5] + d[6] + d[7];
    if (lane == 0)  smem2[2 * wid + 0] = s;   // rows 0..7 total
    if (lane == 16) smem2[2 * wid + 1] = s;   // rows 8..15 total
}

// ---------------- Kernel 1: ragged prefix reduction -------------------------
// grid = B * SPLIT blocks. Block (b, chunk) reduces rows c = chunk, chunk+SPLIT, ...
// over columns l < zero_pos[b]. Writes (sum, sumsq) partial per (b, chunk).
__global__ void tln_reduce_kernel(const float* __restrict__ data,
                                  const int* __restrict__ zero_pos,
                                  float* __restrict__ partials) {
    const int b     = blockIdx.x / SPLIT;
    const int chunk = blockIdx.x % SPLIT;
    const int zp    = zero_pos[b];
    const int tid   = threadIdx.x;

    float s0 = 0.f, s1 = 0.f, q0 = 0.f, q1 = 0.f;
    const int n4   = zp >> 2;      // whole float4s per row (uniform across block)
    const int tail = n4 << 2;

    for (int c = chunk; c < Cn; c += SPLIT) {
        const float*  row  = data + ((size_t)b * Cn + c) * Ln;
        const float4* row4 = (const float4*)row;
        for (int v = tid; v < n4; v += TPB) {
            float4 x = row4[v];                 // global_load_b128
            s0 += x.x + x.z;        s1 += x.y + x.w;
            q0 += x.x * x.x + x.z * x.z;
            q1 += x.y * x.y + x.w * x.w;
        }
        for (int l = tail + tid; l < zp; l += TPB) {  // <=3 tail elements
            float x = row[l];
            s0 += x;  q0 += x * x;
        }
    }

    __shared__ float sm_sum[2 * NWAVES];
    __shared__ float sm_sq [2 * NWAVES];
    const int lane = tid & 31, wid = tid >> 5;
    // All loops reconverged here: EXEC is all-1s for the WMMAs.
    wave_reduce_wmma(s0, s1, sm_sum, lane, wid);
    wave_reduce_wmma(q0, q1, sm_sq,  lane, wid);
    __syncthreads();

    if (tid == 0) {
        float ts = 0.f, tq = 0.f;
        for (int w = 0; w < 2 * NWAVES; ++w) { ts += sm_sum[w]; tq += sm_sq[w]; }
        partials[(b * SPLIT + chunk) * 2 + 0] = ts;
        partials[(b * SPLIT + chunk) * 2 + 1] = tq;
    }
}

// ---------------- Kernel 2: combine partials -> (mean, rstd) per batch ------
__global__ void tln_finalize_kernel(const float* __restrict__ partials,
                                    const int* __restrict__ zero_pos,
                                    float* __restrict__ stats) {
    const int b = blockIdx.x * blockDim.x + threadIdx.x;
    if (b >= Bn) return;
    float s = 0.f, q = 0.f;
    for (int k = 0; k < SPLIT; ++k) {           // fixed order -> deterministic
        s += partials[(b * SPLIT + k) * 2 + 0];
        q += partials[(b * SPLIT + k) * 2 + 1];
    }
    const float cnt  = (float)(Cn * zero_pos[b]);
    const float mean = s / cnt;
    const float var  = (q - s * s / cnt) / (cnt - 1.0f);   // unbiased
    const float stdv = sqrtf(var);
    const float rstd = 1.0f / (stdv + 1.1920928955078125e-7f);  // FLT_EPSILON
    stats[2 * b + 0] = mean;
    stats[2 * b + 1] = rstd;
}

// ---------------- Kernel 3: elementwise normalize (float4) ------------------
__global__ void tln_normalize_kernel(const float* __restrict__ data,
                                     const float* __restrict__ stats,
                                     float* __restrict__ out) {
    const size_t total4 = (size_t)Bn * ((Cn * Ln) / 4);
    const size_t stride = (size_t)gridDim.x * blockDim.x;
    for (size_t i4 = (size_t)blockIdx.x * blockDim.x + threadIdx.x;
         i4 < total4; i4 += stride) {
        const int   b    = (int)(i4 >> 17);     // C*L/4 == 2^17
        const float mean = stats[2 * b + 0];
        const float rstd = stats[2 * b + 1];
        float4 x = ((const float4*)data)[i4];
        float4 y;
        y.x = (x.x - mean) * rstd;
        y.y = (x.y - mean) * rstd;
        y.z = (x.z - mean) * rstd;
        y.w = (x.w - mean) * rstd;
        ((float4*)out)[i4] = y;
    }
}

// ---------------- Kernel 4: bit-copy idxes + zero_pos into output tail ------
__global__ void tln_copytail_kernel(const int* __restrict__ idxes,
                                    const int* __restrict__ zero_pos,
                                    int* __restrict__ out_tail) {
    const int n = Bn * Ln;                      // idxes element count
    const int total = n + Bn;
    const int stride = gridDim.x * blockDim.x;
    for (int j = blockIdx.x * blockDim.x + threadIdx.x; j < total; j += stride) {
        out_tail[j] = (j < n) ? idxes[j] : zero_pos[j - n];
    }
}

extern "C" void kernel_launch(void* const* d_in, const int* in_sizes, int n_in,
                              void* d_out, int out_size, void* d_ws, size_t ws_size,
                              hipStream_t stream) {
    (void)in_sizes; (void)n_in; (void)out_size; (void)ws_size;

    const float* data     = (const float*)d_in[0];   // [B,C,L] fp32
    const int*   idxes    = (const int*)d_in[1];     // [B,L]   int32
    const int*   zero_pos = (const int*)d_in[2];     // [B]     int32

    float* out      = (float*)d_out;
    int*   out_tail = (int*)d_out + (size_t)Bn * Cn * Ln;

    float* partials = (float*)d_ws;                  // B*SPLIT*2 floats
    float* stats    = partials + Bn * SPLIT * 2;     // B*2 floats

    tln_reduce_kernel<<<Bn * SPLIT, TPB, 0, stream>>>(data, zero_pos, partials);
    tln_finalize_kernel<<<1, 64, 0, stream>>>(partials, zero_pos, stats);
    tln_normalize_kernel<<<4096, TPB, 0, stream>>>(data, stats, out);
    tln_copytail_kernel<<<512, TPB, 0, stream>>>(idxes, zero_pos, out_tail);
}